// StructureLearinng_84885733638730
// MI455X (gfx1250) — compile-verified
//
#include <hip/hip_runtime.h>
#include <hip/hip_bf16.h>
#include <math.h>

#define EPSF 1e-6f

typedef float v2f __attribute__((ext_vector_type(2)));
typedef float v8f __attribute__((ext_vector_type(8)));

// ---------------- zero u32 ----------------
__global__ void k_zero_u32(unsigned* __restrict__ p, int n) {
  int i = blockIdx.x * blockDim.x + threadIdx.x;
  if (i < n) p[i] = 0u;
}

// ---------------- a1/a2 = x @ att halves, via f32 WMMA ----------------
// Wave handles 16 rows of x. A-frag (16x4 f32, 2 VGPR): lanes 0-15 -> K=kb,kb+1,
// lanes 16-31 -> K=kb+2,kb+3.  B-frag (4x16): column 0 = att[:D] chunk, column 1
// = att[D:] chunk, other columns zero (built branchlessly with lane masks).
// D-frag (16x16 f32): column n lives in lanes (n, n+16); a1 = column 0, a2 = column 1.
__global__ void k_gemv_att(const float* __restrict__ x,
                           const float* __restrict__ att,
                           float* __restrict__ a1, float* __restrict__ a2,
                           int N, int D) {
  __shared__ float satt[256];                     // 2*D, D <= 128
  for (int i = threadIdx.x; i < 2 * D; i += blockDim.x) satt[i] = att[i];
  __syncthreads();
#if __has_builtin(__builtin_amdgcn_wmma_f32_16x16x4_f32)
  const int lane = threadIdx.x & 31;
  const int wv   = threadIdx.x >> 5;
  const int gw   = blockIdx.x * (blockDim.x >> 5) + wv;
  const int base = gw * 16;
  if (base >= N) return;                          // uniform per wave, EXEC stays full
  const int half = lane >> 4;                     // 0: K pair {0,1}, 1: K pair {2,3}
  const int l16  = lane & 15;
  const float m0 = (l16 == 0) ? 1.f : 0.f;        // column-select masks (branchless B)
  const float m1 = (l16 == 1) ? 1.f : 0.f;
  int rowA = base + l16; if (rowA >= N) rowA = N - 1;
  const float* __restrict__ xrow = x + (size_t)rowA * D;
  v8f acc = {0.f, 0.f, 0.f, 0.f, 0.f, 0.f, 0.f, 0.f};
  for (int kb = 0; kb < D; kb += 4) {
    const int ka = kb + (half << 1);              // even -> 8B aligned
    v2f a = *(const v2f*)(xrow + ka);             // A tile: global_load_b64
    v2f ta = *(const v2f*)(satt + ka);            // att[:D] chunk: ds_load_b64
    v2f tb = *(const v2f*)(satt + D + ka);        // att[D:] chunk: ds_load_b64
    v2f b;
    b.x = m0 * ta.x + m1 * tb.x;
    b.y = m0 * ta.y + m1 * tb.y;
    acc = __builtin_amdgcn_wmma_f32_16x16x4_f32(false, a, false, b,
                                                (short)0, acc, false, false);
  }
  if (l16 <= 1) {
    float* dst = (l16 == 0) ? a1 : a2;
    const int mofs = half ? 8 : 0;
#pragma unroll
    for (int i = 0; i < 8; ++i) {
      int r = base + mofs + i;
      if (r < N) dst[r] = acc[i];
    }
  }
#else
  int r = blockIdx.x * blockDim.x + threadIdx.x;
  if (r >= N) return;
  float s1 = 0.f, s2 = 0.f;
  for (int k = 0; k < D; ++k) {
    float v = x[(size_t)r * D + k];
    s1 += v * satt[k];
    s2 += v * satt[D + k];
  }
  a1[r] = s1; a2[r] = s2;
#endif
}

// ---------------- histogram of destination nodes ----------------
__global__ void k_hist(const int* __restrict__ ei, unsigned* __restrict__ cnt,
                       int E, int M) {
  int e = blockIdx.x * blockDim.x + threadIdx.x;
  if (e >= M) return;
  int c = (e < E) ? ei[(size_t)E + e] : (e - E);   // appended self loops
  atomicAdd(&cnt[c], 1u);
}

// ---------------- block-level exclusive scan (1024 elems / block) ----------------
__global__ void k_scan_block(unsigned* __restrict__ data,
                             unsigned* __restrict__ bsum, int n) {
  __shared__ unsigned sh[256];
  const int t = threadIdx.x, b = blockIdx.x;
  const int baseI = b * 1024 + t * 4;
  unsigned v[4]; unsigned s = 0;
#pragma unroll
  for (int i = 0; i < 4; ++i) {
    int id = baseI + i;
    v[i] = (id < n) ? data[id] : 0u;
    s += v[i];
  }
  sh[t] = s;
  __syncthreads();
  for (int ofs = 1; ofs < 256; ofs <<= 1) {
    unsigned val = sh[t];
    unsigned add = (t >= ofs) ? sh[t - ofs] : 0u;
    __syncthreads();
    sh[t] = val + add;
    __syncthreads();
  }
  unsigned run = (t == 0) ? 0u : sh[t - 1];
#pragma unroll
  for (int i = 0; i < 4; ++i) {
    int id = baseI + i;
    if (id < n) data[id] = run;
    run += v[i];
  }
  if (t == 255) bsum[b] = sh[255];
}

__global__ void k_scan_bsum(unsigned* __restrict__ bsum, int nb,
                            unsigned* __restrict__ off, int N, unsigned M) {
  if (blockIdx.x == 0 && threadIdx.x == 0) {
    unsigned run = 0;
    for (int i = 0; i < nb; ++i) { unsigned v = bsum[i]; bsum[i] = run; run += v; }
    off[N] = M;                                   // bucket end sentinel
  }
}

__global__ void k_scan_add(unsigned* __restrict__ off, const unsigned* __restrict__ bsum,
                           unsigned* __restrict__ cur, int n) {
  int i = blockIdx.x * blockDim.x + threadIdx.x;
  if (i >= n) return;
  unsigned v = off[i] + bsum[i >> 10];
  off[i] = v;
  cur[i] = v;                                     // scatter cursor
}

// ---------------- scatter edges to sorted-by-col positions ----------------
__global__ void k_scatter(const int* __restrict__ ei, const float* __restrict__ ewt,
                          const int* __restrict__ emk,
                          const float* __restrict__ a1, const float* __restrict__ a2,
                          unsigned* __restrict__ cur,
                          float* __restrict__ out_row, float* __restrict__ out_col,
                          float* __restrict__ out_ew,  float* __restrict__ out_em,
                          float* __restrict__ w_s, unsigned* __restrict__ idx_s,
                          int E, int M) {
  int e = blockIdx.x * blockDim.x + threadIdx.x;
  if (e >= M) return;
  int r, c; float we, me;
  if (e < E) { r = ei[e]; c = ei[(size_t)E + e]; we = ewt[e]; me = (float)emk[e]; }
  else       { r = c = e - E; we = 0.f; me = -1.f; }
  float wv = a1[r] + a2[c];
  wv = (wv > 0.f) ? wv : 0.01f * wv;              // leaky_relu, slope 0.01
  wv += we;
  unsigned pos = atomicAdd(&cur[c], 1u);
  out_row[pos] = (float)r;
  out_col[pos] = (float)c;
  out_ew[pos]  = we;
  out_em[pos]  = me;
  w_s[pos]     = wv;
  idx_s[pos]   = (unsigned)e;
}

// ---------------- per-bucket: stabilize (stable sort by orig idx) + softmax stats ----
__global__ void k_bucket(const unsigned* __restrict__ off,
                         unsigned* __restrict__ idx_s, float* __restrict__ w_s,
                         float* __restrict__ out_row, float* __restrict__ out_ew,
                         float* __restrict__ out_em,
                         float* __restrict__ mval, float* __restrict__ ssum, int N) {
  int c = blockIdx.x * blockDim.x + threadIdx.x;
  if (c >= N) return;
  const int s0 = (int)off[c], s1 = (int)off[c + 1];
  // insertion sort by original edge index -> reproduces stable argsort order
  for (int i = s0 + 1; i < s1; ++i) {
    unsigned ki = idx_s[i];
    float kw = w_s[i], kr = out_row[i], ke = out_ew[i], km = out_em[i];
    int j = i - 1;
    while (j >= s0 && idx_s[j] > ki) {
      idx_s[j + 1] = idx_s[j]; w_s[j + 1] = w_s[j]; out_row[j + 1] = out_row[j];
      out_ew[j + 1] = out_ew[j]; out_em[j + 1] = out_em[j];
      --j;
    }
    idx_s[j + 1] = ki; w_s[j + 1] = kw; out_row[j + 1] = kr;
    out_ew[j + 1] = ke; out_em[j + 1] = km;
  }
  float m = -INFINITY;
  for (int i = s0; i < s1; ++i) m = fmaxf(m, w_s[i]);
  float s = 0.f;
  for (int i = s0; i < s1; ++i) s += expf(w_s[i] - m);
  mval[c] = m;
  ssum[c] = s;
}

// ---------------- finalize: softmax prob -> relaxed Bernoulli -> outputs ----------
__global__ void k_finalize(const float* __restrict__ w_s, const float* __restrict__ out_row,
                           const float* __restrict__ out_col,
                           const float* __restrict__ mval, const float* __restrict__ ssum,
                           const float* __restrict__ noise_u, const int* __restrict__ layer,
                           float* __restrict__ out_ew, float* __restrict__ out_em,
                           float* __restrict__ out_ys, float* __restrict__ out_intra,
                           int M) {
  int pos = blockIdx.x * blockDim.x + threadIdx.x;
  if (pos >= M) return;
  const int c = (int)out_col[pos];
  float p = expf(w_s[pos] - mval[c]) / ssum[c];
  p = fminf(fmaxf(p, EPSF), 1.f - EPSF);
  float u = fminf(fmaxf(noise_u[pos], EPSF), 1.f - EPSF);
  float logits = logf(p) - log1pf(-p);
  float noise  = logf(u) - log1pf(-u);
  float z  = (logits + noise) * 2.0f;             // / temperature (0.5)
  float ys = 1.f / (1.f + expf(-z));
  float yh = (ys > 0.5f) ? 1.f : 0.f;
  float y  = (yh - ys) + ys;                      // straight-through, JAX float semantics
  out_ys[pos] = ys;
  float ew = out_ew[pos];
  out_ew[pos] = (ew == 0.f) ? y : ew;
  float em = out_em[pos];
  if (em == 0.f && y == 1.f) out_em[pos] = (float)(layer[0] + 1);
  if ((int)out_row[pos] == c) out_intra[c] = ys;  // exactly one self loop per node
}

extern "C" void kernel_launch(void* const* d_in, const int* in_sizes, int n_in,
                              void* d_out, int out_size, void* d_ws, size_t ws_size,
                              hipStream_t stream) {
  (void)n_in; (void)out_size; (void)ws_size;
  const float* x     = (const float*)d_in[0];
  const int*   ei    = (const int*)d_in[1];     // (2,E) row-major: rows then cols
  const float* ewt   = (const float*)d_in[2];
  const int*   emk   = (const int*)d_in[3];
  const int*   layer = (const int*)d_in[4];
  const float* att   = (const float*)d_in[5];
  const float* nu    = (const float*)d_in[6];

  const int D = in_sizes[5] / 2;                // 128
  const int N = in_sizes[0] / D;                // 100000
  const int E = in_sizes[2];                    // 6400000
  const int M = E + N;

  float* out       = (float*)d_out;
  float* out_row   = out;
  float* out_col   = out + (size_t)M;
  float* out_ew    = out + 2 * (size_t)M;
  float* out_ys    = out + 3 * (size_t)M;
  float* out_em    = out + 4 * (size_t)M;
  float* out_intra = out + 5 * (size_t)M;

  // carve workspace
  char* wsb = (char*)d_ws; size_t o = 0;
  auto carve = [&](size_t bytes) -> void* {
    void* p = wsb + o; o += (bytes + 255) & ~(size_t)255; return p;
  };
  float*    a1    = (float*)carve((size_t)N * 4);
  float*    a2    = (float*)carve((size_t)N * 4);
  unsigned* off   = (unsigned*)carve(((size_t)N + 1) * 4);
  unsigned* cur   = (unsigned*)carve((size_t)N * 4);
  unsigned* bsum  = (unsigned*)carve(4096 * 4);
  float*    mval  = (float*)carve((size_t)N * 4);
  float*    ssum  = (float*)carve((size_t)N * 4);
  float*    w_s   = (float*)carve((size_t)M * 4);
  unsigned* idx_s = (unsigned*)carve((size_t)M * 4);

  const int T = 256;
  k_zero_u32<<<(N + 1 + T - 1) / T, T, 0, stream>>>(off, N + 1);

  const int waves16 = (N + 15) / 16;
  const int gB = (waves16 + 7) / 8;             // 8 waves / 256-thread block
  k_gemv_att<<<gB, 256, 0, stream>>>(x, att, a1, a2, N, D);

  k_hist<<<(M + T - 1) / T, T, 0, stream>>>(ei, off, E, M);

  const int nb = (N + 1023) / 1024;
  k_scan_block<<<nb, 256, 0, stream>>>(off, bsum, N);
  k_scan_bsum<<<1, 32, 0, stream>>>(bsum, nb, off, N, (unsigned)M);
  k_scan_add<<<(N + T - 1) / T, T, 0, stream>>>(off, bsum, cur, N);

  k_scatter<<<(M + T - 1) / T, T, 0, stream>>>(ei, ewt, emk, a1, a2, cur,
                                               out_row, out_col, out_ew, out_em,
                                               w_s, idx_s, E, M);

  k_bucket<<<(N + T - 1) / T, T, 0, stream>>>(off, idx_s, w_s,
                                              out_row, out_ew, out_em,
                                              mval, ssum, N);

  k_finalize<<<(M + T - 1) / T, T, 0, stream>>>(w_s, out_row, out_col, mval, ssum,
                                                nu, layer, out_ew, out_em,
                                                out_ys, out_intra, M);
}